// BinaryTreeCodebook_68118181314715
// MI455X (gfx1250) — compile-verified
//
#include <hip/hip_runtime.h>

// ---------------------------------------------------------------------------
// BinaryTreeCodebook on MI455X (gfx1250, wave32).
//
// Math reduction: bit_k = (r_k . w_k > t_k) with
//   w_k = lev[k,1]-lev[k,0],  t_k = 0.5*(|lev1|^2-|lev0|^2)
//   r_k . w_k = p_k - sum_{j<k} G[j][b_j][k],  p = x @ W  (fp32 GEMM)
// GEMM on V_WMMA_F32_16X16X4_F32 (exact fp32). Bandwidth floor: read x
// (268MB) + write x_q (268MB) -> ~23us @ 23.3TB/s. CDNA5 data movers:
//  - One TDM tensor_load_to_lds DMAs the 129KB constant block (A-fragments +
//    3 nibble decode-LUTs) per block, L2-resident.
//  - global_load_async_to_lds_b64 performs the B-fragment swizzle in the DMA
//    (per-lane LDS dest), tracked on ASYNCcnt, no VGPR staging roundtrip.
//  - Decode uses 3x ds_load_b128 nibble-LUT lookups per float4 (4x fewer DS
//    ops than per-level accumulation), keeping LDS below the HBM floor.
// ---------------------------------------------------------------------------

typedef __attribute__((ext_vector_type(2))) float v2f;
typedef __attribute__((ext_vector_type(4))) float v4f;
typedef __attribute__((ext_vector_type(8))) float v8f;
typedef __attribute__((ext_vector_type(4))) unsigned u32x4;
typedef __attribute__((ext_vector_type(8))) int i32x8;
typedef __attribute__((ext_vector_type(4))) int i32x4;

#define DEPTH 12
#define DIM 512
#define NROW (16 * 8192)              // 131072 vectors
#define NELEM ((long long)NROW * DIM) // 67108864
#define WAVES_PER_BLOCK 8
#define ROWS_PER_BLOCK (16 * WAVES_PER_BLOCK) // 128
#define NBLOCKS (NROW / ROWS_PER_BLOCK)       // 1024

#define LUT_STRIDE 516                 // padded row stride (words): banks 4n..
#define LUT_FLOATS (3 * 16 * LUT_STRIDE) // 24768

// workspace layout (floats) -- [AFRAG | LUT] contiguous: one TDM descriptor
#define WS_AFRAG 0                     // 8192 : W in WMMA A-fragment layout
#define WS_LUT 8192                    // 24768: 3 nibble tables (pad 516)
#define WS_G 32960                     // 288  : G[j][b][k] = dot(lev[j][b], w_k)
#define WS_T 33248                     // 12   : thresholds
#define WS_PART 33280                  // 1024 : per-block loss partials

#define TDM_ELEMS ((8192 + LUT_FLOATS) / 2) // 16480 x 8B = 131840 B

// ---------------------------------------------------------------------------
__global__ __launch_bounds__(512) void btc_prep_kernel(
    const float* __restrict__ lev, float* __restrict__ ws) {
  const int tid = threadIdx.x;

  // A-fragment of W: 16x4 fp32 A layout per chunk c (K=4c..4c+3):
  //   lane L, slot s -> A[m=L%16][k = 4c + s + 2*(L>=16)], rows 12..15 = 0
  for (int i = tid; i < 128 * 32 * 2; i += 512) {
    int c = i >> 6, r = (i >> 1) & 31, s = i & 1;
    int m = r & 15;
    int k = 4 * c + s + ((r >> 4) << 1);
    float v = 0.0f;
    if (m < DEPTH) v = lev[m * 1024 + 512 + k] - lev[m * 1024 + k];
    ws[WS_AFRAG + i] = v;
  }
  // Nibble decode LUTs: LUT[t][n][d] = sum_{k in 0..3, n bit k} w[4t+k][d],
  // with base[d] = sum_k lev[k][0][d] folded into table 0.
  for (int i = tid; i < 3 * 16 * DIM; i += 512) {
    int t = i >> 13, rem = i & 8191;
    int n = rem >> 9, d = rem & 511;
    float s = 0.0f;
    if (t == 0) {
      for (int k = 0; k < DEPTH; ++k) s += lev[k * 1024 + d];
    }
    for (int k = 0; k < 4; ++k) {
      if ((n >> k) & 1) {
        int j = 4 * t + k;
        s += lev[j * 1024 + 512 + d] - lev[j * 1024 + d];
      }
    }
    ws[WS_LUT + t * 16 * LUT_STRIDE + n * LUT_STRIDE + d] = s;
  }
  // Gram table
  if (tid < DEPTH * 2 * DEPTH) {
    int j = tid / (2 * DEPTH), rem = tid % (2 * DEPTH);
    int b = rem / DEPTH, k2 = rem % DEPTH;
    const float* a = lev + j * 1024 + b * 512;
    const float* w1 = lev + k2 * 1024 + 512;
    const float* w0 = lev + k2 * 1024;
    float s = 0.0f;
    for (int d = 0; d < DIM; ++d) s += a[d] * (w1[d] - w0[d]);
    ws[WS_G + tid] = s;
  }
  // thresholds
  if (tid < DEPTH) {
    const float* l0 = lev + tid * 1024;
    const float* l1 = l0 + 512;
    float s = 0.0f;
    for (int d = 0; d < DIM; ++d) s += l1[d] * l1[d] - l0[d] * l0[d];
    ws[WS_T + tid] = 0.5f * s;
  }
}

// ---------------------------------------------------------------------------
__global__ __launch_bounds__(256) void btc_main_kernel(
    const float* __restrict__ x, const float* __restrict__ ws_ro,
    float* __restrict__ ws_rw, float* __restrict__ xq,
    int* __restrict__ out_idx) {
  // LDS: 129KB [Afrag|LUT] + 8*8448B B-staging + tables ~= 200KB / block
  __shared__ __align__(16) float sAL[8192 + LUT_FLOATS];
  __shared__ __align__(16) float sB[WAVES_PER_BLOCK][32 * 66];  // padded stride 66
  __shared__ float sG[288];
  __shared__ float sT[16];
  __shared__ float sRed[WAVES_PER_BLOCK];

  float* sA = sAL;                 // A fragments (128 chunks x 32 x 2)
  float* sLUT = sAL + 8192;        // 3 nibble tables

  const int tid = threadIdx.x;
  const int lane = tid & 31;
  const int wv = tid >> 5;

  // --- Block init: one TDM DMA for the 129KB constant image (wave 0 issues),
  //     plain loads for the small tables (all threads).
  if (wv == 0) {
    unsigned lds_off = (unsigned)(size_t)(void*)&sAL[0];
    unsigned long long ga = (unsigned long long)(const void*)(ws_ro + WS_AFRAG);
    // D# group0: count=1 | lds_addr | global_addr[56:0] | type=2
    u32x4 g0 = {1u, lds_off, (unsigned)ga,
                (unsigned)((ga >> 32) & 0x01FFFFFFu) | (2u << 30)};
    // D# group1: data_size=8B(3); tensor_dim0=tile_dim0=stride0=TDM_ELEMS;
    //            tensor_dim1=1; tile_dim1/2 unused.
    i32x8 g1 = {(int)0x00030000u,                    // mask=0, data_size=3
                (int)((TDM_ELEMS & 0xFFFF) << 16),   // tensor_dim0[15:0]
                (int)((TDM_ELEMS >> 16) | (1 << 16)),// dim0[31:16], dim1=1
                (int)((TDM_ELEMS & 0xFFFF) << 16),   // tile_dim0
                0,                                   // tile_dim1/2 unused
                TDM_ELEMS,                           // stride0[31:0]
                0, 0};
    i32x4 gz4 = {0, 0, 0, 0};
    i32x8 gz8 = {0, 0, 0, 0, 0, 0, 0, 0};
    __builtin_amdgcn_tensor_load_to_lds(g0, g1, gz4, gz4, gz8, 0);
    __builtin_amdgcn_s_wait_tensorcnt(0);
  }
  for (int i = tid; i < 288; i += 256) sG[i] = ws_ro[WS_G + i];
  if (tid < DEPTH) sT[tid] = ws_ro[WS_T + tid];
  __syncthreads();

  const int row0 = blockIdx.x * ROWS_PER_BLOCK + wv * 16;
  const float* xt = x + (size_t)row0 * DIM;
  float* myB = &sB[wv][0];
  const unsigned myB_off = (unsigned)(size_t)(void*)myB;

  // p = x_tile @ W via V_WMMA_F32_16X16X4_F32, K staged in quarters of 128.
  v8f acc = {};
  for (int h = 0; h < 4; ++h) {
    const int kb = h * 128;
    // Async DMA the quarter straight into the B-fragment layout:
    // lane L<16 -> (row=L, K=4c+{0,1}), L>=16 -> (row=L-16, K=4c+{2,3}).
    // Chunk stride 66 words keeps the frag reads bank-conflict-free.
#pragma unroll
    for (int i = 0; i < 16; ++i) {
      const float* gaddr = xt + i * DIM + kb + lane * 4;
      unsigned ldsA = myB_off + (unsigned)(lane * 66 + i * 2) * 4u;
      asm volatile("global_load_async_to_lds_b64 %0, %1, off"
                   :: "v"(ldsA), "v"(gaddr) : "memory");
      asm volatile("global_load_async_to_lds_b64 %0, %1, off offset:8"
                   :: "v"(ldsA + 128u), "v"(gaddr) : "memory");
    }
    asm volatile("s_wait_asynccnt 0x0" ::: "memory");
#pragma unroll 8
    for (int cl = 0; cl < 32; ++cl) {
      int c = h * 32 + cl;
      v2f a = *(const v2f*)(&sA[c * 64 + lane * 2]);
      v2f b = *(const v2f*)(myB + cl * 66 + lane * 2);
      acc = __builtin_amdgcn_wmma_f32_16x16x4_f32(
          /*neg_a=*/false, a, /*neg_b=*/false, b,
          /*c_mod=*/(short)0, acc, /*reuse_a=*/false, /*reuse_b=*/false);
    }
    // Next quarter's async writes are safe: the s_wait_dscnt 0 before the
    // last wmma guarantees this quarter's B-fragment reads have completed.
  }

  // D layout: lane l, vgpr v -> p[row = l%16][level = v + 8*(l>=16)].
  // Replicate so every lane holds all 12 p-values of row (lane%16).
  const int r = lane & 15;
  float p[DEPTH];
#pragma unroll
  for (int v = 0; v < 8; ++v) p[v] = __shfl(acc[v], r, 32);
#pragma unroll
  for (int v = 0; v < 4; ++v) p[8 + v] = __shfl(acc[v], r + 16, 32);

  // 12-step greedy recurrence using the Gram table (lane-parallel over rows).
  float corr[DEPTH];
#pragma unroll
  for (int k = 0; k < DEPTH; ++k) corr[k] = 0.0f;
  int idx = 0;
#pragma unroll
  for (int j = 0; j < DEPTH; ++j) {
    float s = p[j] - corr[j];
    int b = (s > sT[j]) ? 1 : 0;
    idx |= b << j;
    const float* g = &sG[(j * 2 + b) * DEPTH];
#pragma unroll
    for (int k = j + 1; k < DEPTH; ++k) corr[k] += g[k];
  }
  if (lane < 16) out_idx[row0 + lane] = idx;

  // Decode + loss via nibble LUTs: x_q = T0[n0] + T1[n1] + T2[n2].
  // Lane handles row = lane%16, dim half by lane>=16. 3 b128 LDS reads per
  // float4; padded stride 516 spreads nibble values across bank groups.
  const float* T0 = sLUT + ((idx) & 15) * LUT_STRIDE;
  const float* T1 = sLUT + (16 + ((idx >> 4) & 15)) * LUT_STRIDE;
  const float* T2 = sLUT + (32 + ((idx >> 8) & 15)) * LUT_STRIDE;

  const int myrow = row0 + r;
  const int doff = (lane < 16) ? 0 : 256;
  const float* xr = x + (size_t)myrow * DIM + doff;  // L0/L2-hot: just streamed
  float* qr = xq + (size_t)myrow * DIM + doff;
  float lsum = 0.0f;
  for (int i = 0; i < 64; ++i) {
    int d = doff + 4 * i;
    v4f xv = *(const v4f*)(xr + 4 * i);
    v4f q = *(const v4f*)(T0 + d) + *(const v4f*)(T1 + d) + *(const v4f*)(T2 + d);
    *(v4f*)(qr + 4 * i) = q;
    v4f df = q - xv;
    lsum += df.x * df.x + df.y * df.y + df.z * df.z + df.w * df.w;
  }

  // Deterministic block reduction (no float atomics -> replay-stable).
#pragma unroll
  for (int o = 16; o > 0; o >>= 1) lsum += __shfl_xor(lsum, o, 32);
  if (lane == 0) sRed[wv] = lsum;
  __syncthreads();
  if (tid == 0) {
    float t = 0.0f;
    for (int w = 0; w < WAVES_PER_BLOCK; ++w) t += sRed[w];
    ws_rw[WS_PART + blockIdx.x] = t;
  }
}

// ---------------------------------------------------------------------------
__global__ __launch_bounds__(256) void btc_finish_kernel(
    const float* __restrict__ ws, float* __restrict__ loss_out) {
  __shared__ float sRed[8];
  const int tid = threadIdx.x;
  const int lane = tid & 31;
  const int wv = tid >> 5;
  float s = 0.0f;
#pragma unroll
  for (int i = 0; i < 4; ++i) s += ws[WS_PART + tid * 4 + i];
#pragma unroll
  for (int o = 16; o > 0; o >>= 1) s += __shfl_xor(s, o, 32);
  if (lane == 0) sRed[wv] = s;
  __syncthreads();
  if (tid == 0) {
    float t = 0.0f;
    for (int w = 0; w < 8; ++w) t += sRed[w];
    *loss_out = t * (2.0f / (float)NELEM);
  }
}

// ---------------------------------------------------------------------------
extern "C" void kernel_launch(void* const* d_in, const int* in_sizes, int n_in,
                              void* d_out, int out_size, void* d_ws,
                              size_t ws_size, hipStream_t stream) {
  (void)in_sizes; (void)n_in; (void)out_size; (void)ws_size;
  const float* x = (const float*)d_in[0];     // (16, 8192, 512) f32
  const float* lev = (const float*)d_in[1];   // (12, 2, 512) f32
  float* ws = (float*)d_ws;

  // d_out: [x_q_ste (67108864 f32)] [loss (1 f32)] [indices (131072 i32)]
  float* xq = (float*)d_out;
  float* loss = (float*)d_out + (size_t)NELEM;
  int* oidx = (int*)((float*)d_out + (size_t)NELEM + 1);

  btc_prep_kernel<<<1, 512, 0, stream>>>(lev, ws);
  btc_main_kernel<<<NBLOCKS, 256, 0, stream>>>(x, ws, ws, xq, oidx);
  btc_finish_kernel<<<1, 256, 0, stream>>>(ws, loss);
}